// GATModel_36601711296693
// MI455X (gfx1250) — compile-verified
//
#include <hip/hip_runtime.h>
#include <math.h>

#define N_NODES   50000
#define N_EDGES   800000
#define E_TOT     (N_EDGES + N_NODES)   /* 850000 with self loops */
#define IN_DIM    256
#define HID       64
#define OUT_DIM   32
#define NUM_GRAPHS 64
#define NEG_SLOPE 0.2f

typedef float v2f __attribute__((ext_vector_type(2)));
typedef float v8f __attribute__((ext_vector_type(8)));

// ---------------------------------------------------------------------------
// WMMA fp32 GEMM:  D[M,N] = act(A[M,K] @ B[K,N] (+ bias))
// V_WMMA_F32_16X16X4_F32, wave32. Each wave owns a 16 x (16*NTG) output strip:
// the A fragment (one global_load_b64 per lane per k-step) is reused across
// NTG back-to-back WMMAs, cutting A HBM traffic by NTG.
// B is staged in LDS pre-swizzled into per-lane v2f fragments so each WMMA
// needs exactly one ds_load_b64:
//   Bsw[(kstep*2 + kr)*N + n] = { B[4*kstep+kr][n], B[4*kstep+kr+2][n] },
// where kr = lane>>4 selects the K rows that lane holds (A: lanes 0-15 hold
// K=0..1, lanes 16-31 hold K=2..3; C/D: 8 VGPRs, M = v (+8 for lanes>=16)).
// M must be a multiple of 16 (50000 = 3125*16); K*N/2 <= 8192 fragments.
// ---------------------------------------------------------------------------
template <int NTG>
__global__ __launch_bounds__(256) void wmma_gemm_f32_kernel(
    const float* __restrict__ A, const float* __restrict__ B,
    const float* __restrict__ bias, float* __restrict__ D,
    int M, int K, int N, int do_relu)
{
  __shared__ v2f Bsw[8192];                    // 64 KB LDS max (K*N/2 frags)
  const int tid = threadIdx.x;
  const int nfrag = ((K >> 2) << 1) * N;       // K*N/2
  for (int i = tid; i < nfrag; i += 256) {
    const int n  = i % N;
    const int kk = i / N;                      // kstep*2 + kr
    const int kr = kk & 1;
    const int k4 = (kk >> 1) << 2;
    v2f f;
    f.x = B[(k4 + kr) * N + n];
    f.y = B[(k4 + kr + 2) * N + n];
    Bsw[i] = f;
  }
  __syncthreads();

  const int wave    = tid >> 5;
  const int lane    = tid & 31;
  const int ntiles  = N >> 4;
  const int ngroups = ntiles / NTG;            // N-tile groups per row strip
  const int mtiles  = M >> 4;
  const int work    = blockIdx.x * 8 + wave;   // one (mtile, ngroup) per wave
  if (work >= mtiles * ngroups) return;        // wave-uniform exit (EXEC full)
  const int mt = work / ngroups;
  const int ng = work % ngroups;

  const int row   = mt * 16 + (lane & 15);     // A row owned by this lane
  const int kr    = lane >> 4;                 // lanes>=16 hold K rows 2,3
  const int kOff  = kr << 1;
  const int ncol0 = ng * NTG * 16 + (lane & 15);

  v8f acc[NTG];
  #pragma unroll
  for (int t = 0; t < NTG; ++t) acc[t] = (v8f){0.f,0.f,0.f,0.f,0.f,0.f,0.f,0.f};

  const float* Arow = A + (size_t)row * K;
  for (int k = 0; k < K; k += 4) {
    const v2f a = *(const v2f*)(Arow + k + kOff);          // global_load_b64
    const v2f* brow = &Bsw[((((k >> 2) << 1) + kr) * N) + ncol0];
    #pragma unroll
    for (int t = 0; t < NTG; ++t) {
      const v2f b = brow[t * 16];                          // ds_load_b64
      acc[t] = __builtin_amdgcn_wmma_f32_16x16x4_f32(
          /*neg_a=*/false, a, /*neg_b=*/false, b,
          /*c_mod=*/(short)0, acc[t], /*reuse_a=*/false, /*reuse_b=*/false);
    }
  }

  const int mrow0 = mt * 16 + (kr << 3);       // lanes>=16 -> rows M=8..15
  #pragma unroll
  for (int t = 0; t < NTG; ++t) {
    const int ncol = ncol0 + t * 16;
    const float bv = bias ? bias[ncol] : 0.f;
    #pragma unroll
    for (int v = 0; v < 8; ++v) {
      float val = acc[t][v] + bv;
      if (do_relu) val = fmaxf(val, 0.f);
      D[(size_t)(mrow0 + v) * N + ncol] = val;
    }
  }
}

// ---------------------------------------------------------------------------
// Per-(node, head) attention coefficients: a_src = <h, att_src>, a_dst = <h, att_dst>
// ---------------------------------------------------------------------------
__global__ void attn_coef_kernel(const float* __restrict__ hlin,
                                 const float* __restrict__ att_src,
                                 const float* __restrict__ att_dst,
                                 float* __restrict__ asrc, float* __restrict__ adst,
                                 int N, int heads, int C)
{
  int idx = blockIdx.x * blockDim.x + threadIdx.x;
  if (idx >= N * heads) return;
  const int n = idx / heads, h = idx % heads;
  const float* hp = hlin + (size_t)n * heads * C + (size_t)h * C;
  const float* as = att_src + h * C;
  const float* ad = att_dst + h * C;
  float s = 0.f, d = 0.f;
  for (int c = 0; c < C; ++c) { float v = hp[c]; s += v * as[c]; d += v * ad[c]; }
  asrc[idx] = s;
  adst[idx] = d;
}

__global__ void fill_kernel(float* __restrict__ p, float v, int n)
{
  int i = blockIdx.x * blockDim.x + threadIdx.x;
  if (i < n) p[i] = v;
}

// ordered float atomic max via int/uint trick (works with -inf init)
__device__ inline void atomicMaxF32(float* addr, float val)
{
  if (__float_as_int(val) >= 0)
    atomicMax((int*)addr, __float_as_int(val));
  else
    atomicMin((unsigned int*)addr, __float_as_uint(val));
}

__device__ inline void edge_endpoints(const int* ei0, const int* ei1,
                                      int e, int& s, int& d)
{
  if (e < N_EDGES) { s = ei0[e]; d = ei1[e]; }
  else             { s = d = e - N_EDGES; }     // synthesized self loop
}

// leaky-relu logits + per-destination segment max
__global__ void edge_logit_kernel(const int* __restrict__ ei0, const int* __restrict__ ei1,
                                  const float* __restrict__ asrc, const float* __restrict__ adst,
                                  float* __restrict__ logit, float* __restrict__ nmax, int heads)
{
  int e = blockIdx.x * blockDim.x + threadIdx.x;
  if (e >= E_TOT) return;
  int s, d; edge_endpoints(ei0, ei1, e, s, d);
  for (int h = 0; h < heads; ++h) {
    float x = asrc[s * heads + h] + adst[d * heads + h];
    x = (x > 0.f) ? x : NEG_SLOPE * x;
    logit[(size_t)e * heads + h] = x;
    atomicMaxF32(&nmax[d * heads + h], x);
  }
}

// exp(logit - segmax) in place + per-destination denominator
__global__ void edge_exp_kernel(const int* __restrict__ ei0, const int* __restrict__ ei1,
                                float* __restrict__ exbuf, const float* __restrict__ nmax,
                                float* __restrict__ nsum, int heads)
{
  int e = blockIdx.x * blockDim.x + threadIdx.x;
  if (e >= E_TOT) return;
  int s, d; edge_endpoints(ei0, ei1, e, s, d);
  for (int h = 0; h < heads; ++h) {
    size_t idx = (size_t)e * heads + h;
    float ex = __expf(exbuf[idx] - nmax[d * heads + h]);
    exbuf[idx] = ex;
    atomicAdd(&nsum[d * heads + h], ex);
  }
}

// alpha = ex / (denom + eps), written straight into the d_out alpha slot
__global__ void alpha_norm_kernel(const int* __restrict__ ei0, const int* __restrict__ ei1,
                                  const float* __restrict__ exbuf, const float* __restrict__ nsum,
                                  float* __restrict__ alpha, int heads)
{
  int i = blockIdx.x * blockDim.x + threadIdx.x;
  if (i >= E_TOT * heads) return;
  const int e = i / heads, h = i % heads;
  int s, d; edge_endpoints(ei0, ei1, e, s, d);
  alpha[i] = exbuf[i] / (nsum[d * heads + h] + 1e-16f);
}

// out[dst] += alpha * h[src]   (one thread per (edge, channel); coalesced atomics)
__global__ __launch_bounds__(256) void edge_agg_kernel(
    const int* __restrict__ ei0, const int* __restrict__ ei1,
    const float* __restrict__ hlin, const float* __restrict__ alpha,
    float* __restrict__ out, int heads, int C)
{
  const int F   = heads * C;                       // 256 or 32
  const int epb = 256 / F;                         // edges per block
  const int e   = blockIdx.x * epb + (int)(threadIdx.x / F);
  const int f   = threadIdx.x % F;
  if (e >= E_TOT) return;
  int s, d; edge_endpoints(ei0, ei1, e, s, d);
  const int h = f / C;
  const float a = alpha[(size_t)e * heads + h];
  atomicAdd(&out[(size_t)d * F + f], a * hlin[(size_t)s * F + f]);
}

__global__ void bias_act_kernel(float* __restrict__ y, const float* __restrict__ b,
                                int n, int F, int do_relu)
{
  int i = blockIdx.x * blockDim.x + threadIdx.x;
  if (i >= n) return;
  float v = y[i] + b[i % F];
  if (do_relu) v = fmaxf(v, 0.f);
  y[i] = v;
}

// segment sums over batch ids (graph pooling)
__global__ void pool_accum_kernel(const float* __restrict__ h2, const int* __restrict__ batch,
                                  float* __restrict__ sums, float* __restrict__ cnts, int N, int C)
{
  int i = blockIdx.x * blockDim.x + threadIdx.x;
  if (i >= N * C) return;
  const int n = i / C, c = i % C;
  const int g = batch[n];
  atomicAdd(&sums[g * C + c], h2[i]);
  if (c == 0) atomicAdd(&cnts[g], 1.f);
}

// one wave32 per graph: mean + log_softmax over OUT_DIM == warpSize == 32
__global__ void pool_logsoftmax_kernel(const float* __restrict__ sums,
                                       const float* __restrict__ cnts,
                                       float* __restrict__ out, int C)
{
  const int g = blockIdx.x, c = threadIdx.x;
  const float cnt = fmaxf(cnts[g], 1.f);
  float v = sums[g * C + c] / cnt;
  float m = v;
  for (int off = 16; off; off >>= 1) m = fmaxf(m, __shfl_xor(m, off));
  float ex = __expf(v - m), ssum = ex;
  for (int off = 16; off; off >>= 1) ssum += __shfl_xor(ssum, off);
  out[g * C + c] = (v - m) - __logf(ssum);
}

// write the self-loop-augmented edge index (as float) into both output slots
__global__ void write_ei_kernel(const int* __restrict__ ei0, const int* __restrict__ ei1,
                                float* __restrict__ o1, float* __restrict__ o2)
{
  int e = blockIdx.x * blockDim.x + threadIdx.x;
  if (e >= E_TOT) return;
  float s, d;
  if (e < N_EDGES) { s = (float)ei0[e]; d = (float)ei1[e]; }
  else             { s = d = (float)(e - N_EDGES); }
  o1[e] = s; o1[E_TOT + e] = d;
  o2[e] = s; o2[E_TOT + e] = d;
}

// ---------------------------------------------------------------------------
extern "C" void kernel_launch(void* const* d_in, const int* in_sizes, int n_in,
                              void* d_out, int out_size, void* d_ws, size_t ws_size,
                              hipStream_t stream)
{
  (void)in_sizes; (void)n_in; (void)out_size; (void)ws_size;

  const float* x        = (const float*)d_in[0];
  const int*   ei       = (const int*)  d_in[1];
  const int*   batch    = (const int*)  d_in[2];
  const float* W_red    = (const float*)d_in[3];
  const float* b_red    = (const float*)d_in[4];
  const float* W1       = (const float*)d_in[5];
  const float* att_src1 = (const float*)d_in[6];
  const float* att_dst1 = (const float*)d_in[7];
  const float* b1       = (const float*)d_in[8];
  const float* W2       = (const float*)d_in[9];
  const float* att_src2 = (const float*)d_in[10];
  const float* att_dst2 = (const float*)d_in[11];
  const float* b2       = (const float*)d_in[12];
  const int* ei0 = ei;
  const int* ei1 = ei + N_EDGES;

  // ---- output layout: pooled | ei | alpha1 | ei | alpha2 (all as float) ----
  float* out        = (float*)d_out;
  float* out_pool   = out;                          // 64*32
  float* out_ei1    = out_pool   + NUM_GRAPHS * OUT_DIM;
  float* out_alpha1 = out_ei1    + 2 * (size_t)E_TOT;
  float* out_ei2    = out_alpha1 + 4 * (size_t)E_TOT;
  float* out_alpha2 = out_ei2    + 2 * (size_t)E_TOT;

  // ---- workspace carve-up ----
  float* w = (float*)d_ws;
  size_t off = 0;
  float* h0    = w + off; off += (size_t)N_NODES * HID;       // relu(x@W_red+b)
  float* h1lin = w + off; off += (size_t)N_NODES * 4 * HID;   // h0@W1
  float* asrc1 = w + off; off += (size_t)N_NODES * 4;
  float* adst1 = w + off; off += (size_t)N_NODES * 4;
  float* nmax1 = w + off; off += (size_t)N_NODES * 4;
  float* nsum1 = w + off; off += (size_t)N_NODES * 4;
  float* ex1   = w + off; off += (size_t)E_TOT   * 4;         // logits -> exp
  float* out1  = w + off; off += (size_t)N_NODES * 4 * HID;   // aggregated -> h1
  float* h2lin = w + off; off += (size_t)N_NODES * OUT_DIM;   // h1@W2
  float* asrc2 = w + off; off += (size_t)N_NODES;
  float* adst2 = w + off; off += (size_t)N_NODES;
  float* nmax2 = w + off; off += (size_t)N_NODES;
  float* nsum2 = w + off; off += (size_t)N_NODES;
  float* ex2   = w + off; off += (size_t)E_TOT;
  float* out2  = w + off; off += (size_t)N_NODES * OUT_DIM;   // aggregated -> h2
  float* psum  = w + off; off += (size_t)NUM_GRAPHS * OUT_DIM;
  float* pcnt  = w + off; off += (size_t)NUM_GRAPHS;

  const int BT = 256;
  auto cdiv = [](long long a, long long b) { return (int)((a + b - 1) / b); };

  // ---- layer 0: h0 = relu(x @ W_red + b_red)  [50000,256]x[256,64] ----
  {
    int work = (N_NODES / 16) * ((HID / 16) / 4);             // NTG=4 -> 1 group
    wmma_gemm_f32_kernel<4><<<cdiv(work, 8), BT, 0, stream>>>(
        x, W_red, b_red, h0, N_NODES, IN_DIM, HID, 1);
  }

  // ---- GAT layer 1 (4 heads, C=64) ----
  {
    int work = (N_NODES / 16) * (((4 * HID) / 16) / 4);       // NTG=4 -> 4 groups
    wmma_gemm_f32_kernel<4><<<cdiv(work, 8), BT, 0, stream>>>(
        h0, W1, nullptr, h1lin, N_NODES, HID, 4 * HID, 0);
  }
  attn_coef_kernel<<<cdiv((long long)N_NODES * 4, BT), BT, 0, stream>>>(
      h1lin, att_src1, att_dst1, asrc1, adst1, N_NODES, 4, HID);
  fill_kernel<<<cdiv((long long)N_NODES * 4, BT), BT, 0, stream>>>(
      nmax1, -INFINITY, N_NODES * 4);
  hipMemsetAsync(nsum1, 0, (size_t)N_NODES * 4 * sizeof(float), stream);
  hipMemsetAsync(out1,  0, (size_t)N_NODES * 4 * HID * sizeof(float), stream);
  edge_logit_kernel<<<cdiv(E_TOT, BT), BT, 0, stream>>>(
      ei0, ei1, asrc1, adst1, ex1, nmax1, 4);
  edge_exp_kernel<<<cdiv(E_TOT, BT), BT, 0, stream>>>(
      ei0, ei1, ex1, nmax1, nsum1, 4);
  alpha_norm_kernel<<<cdiv((long long)E_TOT * 4, BT), BT, 0, stream>>>(
      ei0, ei1, ex1, nsum1, out_alpha1, 4);
  edge_agg_kernel<<<E_TOT, BT, 0, stream>>>(                  // 256 ch -> 1 edge/block
      ei0, ei1, h1lin, out_alpha1, out1, 4, HID);
  bias_act_kernel<<<cdiv((long long)N_NODES * 4 * HID, BT), BT, 0, stream>>>(
      out1, b1, N_NODES * 4 * HID, 4 * HID, 1);               // h1 = relu(out1 + b1)

  // ---- GAT layer 2 (1 head, C=32) ----
  {
    int work = (N_NODES / 16) * ((OUT_DIM / 16) / 2);         // NTG=2 -> 1 group
    wmma_gemm_f32_kernel<2><<<cdiv(work, 8), BT, 0, stream>>>(
        out1, W2, nullptr, h2lin, N_NODES, 4 * HID, OUT_DIM, 0);
  }
  attn_coef_kernel<<<cdiv(N_NODES, BT), BT, 0, stream>>>(
      h2lin, att_src2, att_dst2, asrc2, adst2, N_NODES, 1, OUT_DIM);
  fill_kernel<<<cdiv(N_NODES, BT), BT, 0, stream>>>(nmax2, -INFINITY, N_NODES);
  hipMemsetAsync(nsum2, 0, (size_t)N_NODES * sizeof(float), stream);
  hipMemsetAsync(out2,  0, (size_t)N_NODES * OUT_DIM * sizeof(float), stream);
  hipMemsetAsync(psum,  0, (size_t)NUM_GRAPHS * OUT_DIM * sizeof(float), stream);
  hipMemsetAsync(pcnt,  0, (size_t)NUM_GRAPHS * sizeof(float), stream);
  edge_logit_kernel<<<cdiv(E_TOT, BT), BT, 0, stream>>>(
      ei0, ei1, asrc2, adst2, ex2, nmax2, 1);
  edge_exp_kernel<<<cdiv(E_TOT, BT), BT, 0, stream>>>(
      ei0, ei1, ex2, nmax2, nsum2, 1);
  alpha_norm_kernel<<<cdiv(E_TOT, BT), BT, 0, stream>>>(
      ei0, ei1, ex2, nsum2, out_alpha2, 1);
  edge_agg_kernel<<<cdiv(E_TOT, 8), BT, 0, stream>>>(         // 32 ch -> 8 edges/block
      ei0, ei1, h2lin, out_alpha2, out2, 1, OUT_DIM);
  bias_act_kernel<<<cdiv((long long)N_NODES * OUT_DIM, BT), BT, 0, stream>>>(
      out2, b2, N_NODES * OUT_DIM, OUT_DIM, 0);               // h2 = out2 + b2

  // ---- pooling + log_softmax ----
  pool_accum_kernel<<<cdiv((long long)N_NODES * OUT_DIM, BT), BT, 0, stream>>>(
      out2, batch, psum, pcnt, N_NODES, OUT_DIM);
  pool_logsoftmax_kernel<<<NUM_GRAPHS, 32, 0, stream>>>(psum, pcnt, out_pool, OUT_DIM);

  // ---- auxiliary outputs: edge index (with self loops), twice ----
  write_ei_kernel<<<cdiv(E_TOT, BT), BT, 0, stream>>>(ei0, ei1, out_ei1, out_ei2);
}